// qpair_34634616275060
// MI455X (gfx1250) — compile-verified
//
#include <hip/hip_runtime.h>
#include <hip/hip_bf16.h>

typedef float v2f __attribute__((ext_vector_type(2)));
typedef float v8f __attribute__((ext_vector_type(8)));

#define N_AGENTS 128
#define H_DIM    128
#define IN_DIM   132          // state(128) + i + j + a_i + a_j
#define PAIRS    8128         // 128*127/2
#define WPB      8            // waves (pairs) per workgroup

__global__ void qpair_zero_kernel(float* __restrict__ out, int n) {
    int t = blockIdx.x * blockDim.x + threadIdx.x;
    if (t < n) out[t] = 0.0f;
}

__global__ __launch_bounds__(WPB * 32)
void qpair_mlp_kernel(const float* __restrict__ state,
                      const float* __restrict__ actions,
                      const float* __restrict__ W1, const float* __restrict__ b1,
                      const float* __restrict__ W2, const float* __restrict__ b2,
                      const float* __restrict__ W3, const float* __restrict__ b3,
                      float* __restrict__ lam)
{
    // per-wave LDS slots: xs holds x (132, padded to 136 -> 16B aligned slots),
    // hs holds the hidden vector between layers.
    __shared__ float xs[WPB][136];
    __shared__ float hs[WPB][128];

    const int lane = threadIdx.x & 31;
    const int wave = threadIdx.x >> 5;
    const int p    = blockIdx.x * WPB + wave;

    // decode pair index p -> (i, j), i < j (wave-uniform scalar loop)
    int i = 0, rem = p;
    while (rem >= (N_AGENTS - 1 - i)) { rem -= (N_AGENTS - 1 - i); ++i; }
    const int j = i + 1 + rem;

    // stage x = [state, i, j, a_i, a_j] into LDS
    const float4 sv = ((const float4*)state)[lane];
    *(float4*)&xs[wave][lane * 4] = sv;
    if (lane == 0) {
        xs[wave][128] = (float)i;
        xs[wave][129] = (float)j;
        xs[wave][130] = actions[i];
        xs[wave][131] = actions[j];
    }
    __syncthreads();

    const int lo   = lane & 15;          // column within 16-wide N chunk
    const int koff = (lane < 16) ? 0 : 2; // A/B K sub-row held by this lane half

    // ---------------- layer 1: h1 = relu(x @ W1 + b1), K=132 ----------------
    {
        const float* w1p = W1 + (size_t)p * IN_DIM * H_DIM;
        const float* b1p = b1 + (size_t)p * H_DIM;
        v8f acc[8];
#pragma unroll
        for (int c = 0; c < 8; ++c) {
            float bv = b1p[c * 16 + lo];   // C: every row = bias[n]
#pragma unroll
            for (int r = 0; r < 8; ++r) acc[c][r] = bv;
        }
        for (int kk = 0; kk < IN_DIM / 4; ++kk) {
            const int k = kk * 4 + koff;
            // A 16x4, all rows identical = x[k..k+3]
            v2f a = *(const v2f*)&xs[wave][k];
#pragma unroll
            for (int c = 0; c < 8; ++c) {
                v2f bm;
                bm.x = w1p[(size_t)k       * H_DIM + c * 16 + lo];
                bm.y = w1p[(size_t)(k + 1) * H_DIM + c * 16 + lo];
                acc[c] = __builtin_amdgcn_wmma_f32_16x16x4_f32(
                    false, a, false, bm, (short)0, acc[c], false, false);
            }
        }
        if (lane < 16) {
#pragma unroll
            for (int c = 0; c < 8; ++c)
                hs[wave][c * 16 + lane] = fmaxf(acc[c][0], 0.0f);
        }
    }
    __syncthreads();

    // ---------------- layer 2: h2 = relu(h1 @ W2 + b2), K=128 ----------------
    {
        const float* w2p = W2 + (size_t)p * H_DIM * H_DIM;
        const float* b2p = b2 + (size_t)p * H_DIM;
        v8f acc[8];
#pragma unroll
        for (int c = 0; c < 8; ++c) {
            float bv = b2p[c * 16 + lo];
#pragma unroll
            for (int r = 0; r < 8; ++r) acc[c][r] = bv;
        }
        for (int kk = 0; kk < H_DIM / 4; ++kk) {
            const int k = kk * 4 + koff;
            v2f a = *(const v2f*)&hs[wave][k];
#pragma unroll
            for (int c = 0; c < 8; ++c) {
                v2f bm;
                bm.x = w2p[(size_t)k       * H_DIM + c * 16 + lo];
                bm.y = w2p[(size_t)(k + 1) * H_DIM + c * 16 + lo];
                acc[c] = __builtin_amdgcn_wmma_f32_16x16x4_f32(
                    false, a, false, bm, (short)0, acc[c], false, false);
            }
        }
        // h2 overwrites the x slot (x no longer needed)
        if (lane < 16) {
#pragma unroll
            for (int c = 0; c < 8; ++c)
                xs[wave][c * 16 + lane] = fmaxf(acc[c][0], 0.0f);
        }
    }
    __syncthreads();

    // -------- layer 3: z = h2 @ W3 + b3 (128x2), softmax, scatter-add --------
    {
        const float* w3p = W3 + (size_t)p * H_DIM * 2;
        float z0 = 0.0f, z1 = 0.0f;
#pragma unroll
        for (int t = 0; t < 4; ++t) {
            const int k = t * 32 + lane;
            const float h = xs[wave][k];
            const float2 w = ((const float2*)w3p)[k];
            z0 += h * w.x;
            z1 += h * w.y;
        }
#pragma unroll
        for (int off = 16; off >= 1; off >>= 1) {
            z0 += __shfl_xor(z0, off, 32);
            z1 += __shfl_xor(z1, off, 32);
        }
        if (lane == 0) {
            z0 += b3[p * 2 + 0];
            z1 += b3[p * 2 + 1];
            const float m  = fmaxf(z0, z1);
            const float e0 = __expf(z0 - m);
            const float e1 = __expf(z1 - m);
            const float inv = 1.0f / (e0 + e1);
            const float g0 = e0 * inv, g1 = e1 * inv;
            atomicAdd(&lam[i * 2 + 0], g0);
            atomicAdd(&lam[i * 2 + 1], g1);
            atomicAdd(&lam[j * 2 + 0], g0);
            atomicAdd(&lam[j * 2 + 1], g1);
        }
    }
}

extern "C" void kernel_launch(void* const* d_in, const int* in_sizes, int n_in,
                              void* d_out, int out_size, void* d_ws, size_t ws_size,
                              hipStream_t stream) {
    const float* state   = (const float*)d_in[0];
    const float* actions = (const float*)d_in[1];
    const float* W1 = (const float*)d_in[2];
    const float* b1 = (const float*)d_in[3];
    const float* W2 = (const float*)d_in[4];
    const float* b2 = (const float*)d_in[5];
    const float* W3 = (const float*)d_in[6];
    const float* b3 = (const float*)d_in[7];
    float* lam = (float*)d_out;

    // output accumulates via atomics -> zero it first (graph-capture safe)
    hipLaunchKernelGGL(qpair_zero_kernel, dim3(1), dim3(256), 0, stream, lam, out_size);
    hipLaunchKernelGGL(qpair_mlp_kernel, dim3(PAIRS / WPB), dim3(WPB * 32), 0, stream,
                       state, actions, W1, b1, W2, b2, W3, b3, lam);
}